// GenerateProposalsOp_32976758899348
// MI455X (gfx1250) — compile-verified
//
#include <hip/hip_runtime.h>
#include <stdint.h>

#define Bn 8
#define An 15
#define Hn 100
#define Wn 168
#define HWn (Hn*Wn)              // 16800
#define NPER (An*Hn*Wn)          // 252000
#define PRE 2000
#define POST 1000
#define CAP 4096
#define BINS 4096
#define WPR 63                   // 32-bit mask words per row (ceil(2000/32))
#define MROW 64                  // padded row stride
#define ROWS_PER_BLK (PRE/16)    // 125
#define NMS_T 0.7f
#define XFORM_CLIP 4.135166556742356f   // log(1000/16)

typedef unsigned int u32;
typedef unsigned long long u64;
typedef u32 u32x4 __attribute__((ext_vector_type(4)));
typedef int  i32x8 __attribute__((ext_vector_type(8)));
typedef int  i32x4 __attribute__((ext_vector_type(4)));

__device__ __forceinline__ u32 ford(float x){
  u32 s = __float_as_uint(x);
  return (s & 0x80000000u) ? ~s : (s | 0x80000000u);   // order-preserving
}
__device__ __forceinline__ float funord(u32 u){
  return (u & 0x80000000u) ? __uint_as_float(u & 0x7FFFFFFFu) : __uint_as_float(~u);
}

// ---------- K0: zero scratch (hist + tbin + gcnt region) ----------
__global__ void gp_zero(u32* __restrict__ p, int n){
  int i = blockIdx.x*blockDim.x + threadIdx.x;
  if (i < n) p[i] = 0u;
}

// ---------- K1: 12-bit histogram of ordered score bits ----------
__global__ void gp_hist(const float* __restrict__ scores, u32* __restrict__ hist){
  int i = blockIdx.x*blockDim.x + threadIdx.x;
  if (i >= Bn*NPER) return;
  __builtin_prefetch(scores + i + 8192, 0, 0);   // global_prefetch_b8
  int b = i / NPER;
  u32 u = ford(scores[i]);
  atomicAdd(&hist[b*BINS + (u >> 20)], 1u);
}

// ---------- K2: find threshold bin (2000th largest) per image ----------
__global__ void gp_scan(const u32* __restrict__ hist, u32* __restrict__ tbin){
  int b = threadIdx.x;
  if (b >= Bn) return;
  u32 c = 0; int t = 0;
  for (int bin = BINS-1; bin >= 0; --bin){
    c += hist[b*BINS + bin];
    if (c >= PRE){ t = bin; break; }
  }
  tbin[b] = (u32)t;
}

// ---------- K3: gather candidate 64-bit keys ----------
__global__ void gp_gather(const float* __restrict__ scores, const u32* __restrict__ tbin,
                          u32* __restrict__ gcnt, u64* __restrict__ gkeys){
  int i = blockIdx.x*blockDim.x + threadIdx.x;
  if (i >= Bn*NPER) return;
  int b = i / NPER;
  int r = i - b*NPER;                 // (a,h,w) in input memory order
  u32 u = ford(scores[i]);
  if ((u >> 20) >= tbin[b]){
    int a  = r / HWn;
    int hw = r - a*HWn;
    u32 f = (u32)(hw*An + a);         // transposed flat index (h,w,a)
    u32 pos = atomicAdd(&gcnt[b], 1u);
    if (pos < CAP)
      gkeys[(size_t)b*CAP + pos] = ((u64)u << 32) | (u32)(~f);  // desc score, asc index
  }
}

// ---------- K4: per-image bitonic sort (desc) of candidates, emit top-2000 ----------
__global__ __launch_bounds__(1024) void gp_sortsel(const u64* __restrict__ gkeys,
                                                   const u32* __restrict__ gcnt,
                                                   int* __restrict__ topi, float* __restrict__ tops){
  __shared__ u64 sk[CAP];
  int b = blockIdx.x, t = threadIdx.x;
  int n = (int)gcnt[b]; if (n > CAP) n = CAP;
  for (int i = t; i < CAP; i += 1024) sk[i] = (i < n) ? gkeys[(size_t)b*CAP + i] : 0ull;
  __syncthreads();
  for (int k = 2; k <= CAP; k <<= 1){
    for (int j = k >> 1; j > 0; j >>= 1){
      #pragma unroll
      for (int p = 0; p < CAP/1024; ++p){
        int i = t + p*1024;
        int l = i ^ j;
        if (l > i){
          u64 a = sk[i], c = sk[l];
          bool dsc = ((i & k) == 0);          // descending overall
          if ((a < c) == dsc){ sk[i] = c; sk[l] = a; }
        }
      }
      __syncthreads();
    }
  }
  for (int j = t; j < PRE; j += 1024){
    u64 kk = sk[j];
    topi[b*PRE + j] = (int)(~((u32)kk));
    tops[b*PRE + j] = funord((u32)(kk >> 32));
  }
}

// ---------- K5: bbox decode + clip + valid ----------
__global__ void gp_transform(const int* __restrict__ topi, const float* __restrict__ deltas,
                             const float* __restrict__ iminfo, const float* __restrict__ anchors,
                             float* __restrict__ props, int* __restrict__ valid){
  int i = blockIdx.x*blockDim.x + threadIdx.x;
  if (i >= Bn*PRE) return;
  int b = i / PRE;
  int f = topi[i];
  int a = f % An;
  int s = f / An;
  int w = s % Wn;
  int h = s / Wn;
  float sx = (float)w * 8.0f, sy = (float)h * 8.0f;
  float ax1 = anchors[a*4+0] + sx, ay1 = anchors[a*4+1] + sy;
  float ax2 = anchors[a*4+2] + sx, ay2 = anchors[a*4+3] + sy;
  float aw = ax2 - ax1 + 1.0f, ah = ay2 - ay1 + 1.0f;
  float cx = ax1 + 0.5f*aw,    cy = ay1 + 0.5f*ah;
  size_t dbase = (((size_t)(b*4*An + 4*a)) * Hn + h) * Wn + w;
  float dx = deltas[dbase];
  float dy = deltas[dbase + (size_t)HWn];
  float dw = fminf(deltas[dbase + 2*(size_t)HWn], XFORM_CLIP);
  float dh = fminf(deltas[dbase + 3*(size_t)HWn], XFORM_CLIP);
  float pcx = dx*aw + cx, pcy = dy*ah + cy;
  float pw = expf(dw)*aw, ph = expf(dh)*ah;
  float x1 = pcx - 0.5f*pw,        y1 = pcy - 0.5f*ph;
  float x2 = pcx + 0.5f*pw - 1.0f, y2 = pcy + 0.5f*ph - 1.0f;
  float imh = iminfo[b*3+0], imw = iminfo[b*3+1], imsc = iminfo[b*3+2];
  float wmax = imw - 1.0f, hmax = imh - 1.0f;
  x1 = fminf(fmaxf(x1, 0.0f), wmax);
  y1 = fminf(fmaxf(y1, 0.0f), hmax);
  x2 = fminf(fmaxf(x2, 0.0f), wmax);
  y2 = fminf(fmaxf(y2, 0.0f), hmax);
  float bw = x2 - x1 + 1.0f, bh = y2 - y1 + 1.0f;
  float ms = 0.0f * imsc;   // MIN_SIZE = 0
  int v = (bw >= ms) && (bh >= ms) && (x1 + bw*0.5f < imw) && (y1 + bh*0.5f < imh);
  ((float4*)props)[i] = make_float4(x1, y1, x2, y2);
  valid[i] = v;
}

// ---------- K6: IoU suppression bitmask; boxes staged to LDS via TDM ----------
__global__ __launch_bounds__(256) void gp_mask(const float* __restrict__ props, u32* __restrict__ mask){
  __shared__ float4 sbox[PRE];                       // 32000 B tile
  int img = blockIdx.x >> 4;
  int blk = blockIdx.x & 15;
  const float* src = props + (size_t)img * PRE * 4;

#if __has_builtin(__builtin_amdgcn_tensor_load_to_lds)
  if (threadIdx.x < 32){                             // wave 0 only (wave-uniform guard)
    unsigned long long ga = (unsigned long long)(uintptr_t)src;
    u32 lds = (u32)(uintptr_t)&sbox[0];              // low 32b of flat addr == LDS byte offset (LDS aperture)
    // D# group0: count=1, lds_addr, 57-bit global_addr, type=2 (bits 127:126)
    u32x4 g0 = { 1u, lds, (u32)ga, ((u32)((ga >> 32) & 0x01FFFFFFu)) | 0x80000000u };
    // D# group1: 1-D tile: data_size=4B, tensor_dim0=tile_dim0=8000, dim1=1, strides=8000
    i32x8 g1 = {
      (int)(2u << 16),                               // data_size = 2 (4 bytes)
      (int)((u32)(PRE*4) << 16),                     // tensor_dim0 lo16 @ bits[63:48]
      (int)(((u32)(PRE*4) >> 16) | (1u << 16)),      // tensor_dim0 hi16 | tensor_dim1 lo16
      (int)((u32)(PRE*4) << 16),                     // tile_dim0 @ bits[127:112]
      1,                                             // tile_dim1 = 1
      (int)(PRE*4),                                  // tensor_dim0_stride lo32
      (int)((u32)(PRE*4) << 16),                     // stride0 hi16(=0) | tensor_dim1_stride lo16
      0
    };
    i32x4 gz4 = {0,0,0,0};
    i32x8 gz8 = {0,0,0,0,0,0,0,0};
    // 6-arg form (amdgpu-toolchain / clang-23): (g0, g1, g2, g3, g4, cpol)
    __builtin_amdgcn_tensor_load_to_lds(g0, g1, gz4, gz4, gz8, 0);
    __builtin_amdgcn_s_wait_tensorcnt(0);            // s_wait_tensorcnt 0
  }
#else
  for (int i = threadIdx.x; i < PRE; i += 256) sbox[i] = ((const float4*)src)[i];
#endif
  __syncthreads();

  const int nt = ROWS_PER_BLK * WPR;                 // 7875 row-word tasks per block
  for (int t = threadIdx.x; t < nt; t += 256){
    int i   = blk*ROWS_PER_BLK + t / WPR;
    int wrd = t % WPR;
    float4 bi = sbox[i];
    float ai = (bi.z - bi.x + 1.0f) * (bi.w - bi.y + 1.0f);
    u32 bits = 0u;
    int j0 = wrd << 5;
    #pragma unroll
    for (int k = 0; k < 32; ++k){
      int j = j0 + k;
      if (j < PRE){
        float4 bj = sbox[j];
        float aj = (bj.z - bj.x + 1.0f) * (bj.w - bj.y + 1.0f);
        float iw = fmaxf(fminf(bi.z, bj.z) - fmaxf(bi.x, bj.x) + 1.0f, 0.0f);
        float ih = fmaxf(fminf(bi.w, bj.w) - fmaxf(bi.y, bj.y) + 1.0f, 0.0f);
        float inter = iw * ih;
        float iou = inter / (ai + aj - inter);
        if (iou > NMS_T) bits |= (1u << k);
      }
    }
    mask[((size_t)(img*PRE + i)) * MROW + wrd] = bits;
  }
}

// ---------- K7: serial greedy NMS scan (matches lax.scan semantics) ----------
__global__ __launch_bounds__(64) void gp_nmsscan(const int* __restrict__ valid,
                                                 const u32* __restrict__ mask,
                                                 int* __restrict__ keep){
  __shared__ u32 remv[WPR];
  __shared__ int cur;
  int b = blockIdx.x, t = threadIdx.x;
  if (t < WPR) remv[t] = 0u;
  __syncthreads();
  for (int i = 0; i < PRE; ++i){
    if (t == 0){
      int sup = (remv[i >> 5] >> (i & 31)) & 1;
      cur = (valid[b*PRE + i] && !sup) ? 1 : 0;
      keep[b*PRE + i] = cur;
    }
    __syncthreads();
    if (cur && t < WPR) remv[t] |= mask[((size_t)(b*PRE + i)) * MROW + t];
    __syncthreads();
  }
}

// ---------- K8: stable compaction + output (rois 8000x5, then probs 8000) ----------
__global__ __launch_bounds__(1024) void gp_output(const float* __restrict__ props,
                                                  const float* __restrict__ tops,
                                                  const int* __restrict__ keep,
                                                  float* __restrict__ out){
  int b = blockIdx.x, t = threadIdx.x;
  float* rois  = out;
  float* probs = out + (size_t)Bn*POST*5;
  for (int r = t; r < POST; r += 1024){
    size_t R = (size_t)b*POST + r;
    rois[R*5+0] = (float)b;
    rois[R*5+1] = 0.0f; rois[R*5+2] = 0.0f; rois[R*5+3] = 0.0f; rois[R*5+4] = 0.0f;
    probs[R] = 0.0f;
  }
  __threadfence_block();
  __syncthreads();
  __shared__ int part[1024];
  int i0 = 2*t, i1 = 2*t + 1;
  int f0 = (i0 < PRE) ? keep[b*PRE + i0] : 0;
  int f1 = (i1 < PRE) ? keep[b*PRE + i1] : 0;
  part[t] = f0 + f1;
  __syncthreads();
  for (int off = 1; off < 1024; off <<= 1){
    int v = (t >= off) ? part[t - off] : 0;
    __syncthreads();
    part[t] += v;
    __syncthreads();
  }
  int base = part[t] - (f0 + f1);
  int p0 = base + f0;
  int p1 = base + f0 + f1;
  if (f0 && p0 <= POST){
    size_t R = (size_t)b*POST + (p0 - 1);
    const float* bx = props + ((size_t)(b*PRE + i0))*4;
    rois[R*5+1] = bx[0]; rois[R*5+2] = bx[1]; rois[R*5+3] = bx[2]; rois[R*5+4] = bx[3];
    probs[R] = tops[b*PRE + i0];
  }
  if (f1 && p1 <= POST){
    size_t R = (size_t)b*POST + (p1 - 1);
    const float* bx = props + ((size_t)(b*PRE + i1))*4;
    rois[R*5+1] = bx[0]; rois[R*5+2] = bx[1]; rois[R*5+3] = bx[2]; rois[R*5+4] = bx[3];
    probs[R] = tops[b*PRE + i1];
  }
}

extern "C" void kernel_launch(void* const* d_in, const int* in_sizes, int n_in,
                              void* d_out, int out_size, void* d_ws, size_t ws_size,
                              hipStream_t stream){
  (void)in_sizes; (void)n_in; (void)out_size; (void)ws_size;
  const float* scores  = (const float*)d_in[0];   // (8,15,100,168)
  const float* deltas  = (const float*)d_in[1];   // (8,60,100,168)
  const float* iminfo  = (const float*)d_in[2];   // (8,3)
  const float* anchors = (const float*)d_in[3];   // (15,4)
  float* out = (float*)d_out;                     // 40000 rois + 8000 probs

  char* basep = (char*)d_ws;
  size_t o = 0;
  auto alloc = [&](size_t bytes)->char*{
    char* r = basep + o; o += (bytes + 255) & ~(size_t)255; return r;
  };
  u32*   hist  = (u32*)  alloc((size_t)Bn*BINS*sizeof(u32));
  u32*   tbin  = (u32*)  alloc((size_t)Bn*sizeof(u32));
  u32*   gcnt  = (u32*)  alloc((size_t)Bn*sizeof(u32));
  u64*   gkeys = (u64*)  alloc((size_t)Bn*CAP*sizeof(u64));
  int*   topi  = (int*)  alloc((size_t)Bn*PRE*sizeof(int));
  float* tops  = (float*)alloc((size_t)Bn*PRE*sizeof(float));
  float* props = (float*)alloc((size_t)Bn*PRE*4*sizeof(float));
  int*   valid = (int*)  alloc((size_t)Bn*PRE*sizeof(int));
  u32*   maskm = (u32*)  alloc((size_t)Bn*PRE*MROW*sizeof(u32));
  int*   keep  = (int*)  alloc((size_t)Bn*PRE*sizeof(int));

  int zn = (int)(((char*)gkeys - (char*)hist) / 4);   // hist+tbin+gcnt region
  gp_zero<<<(zn + 255)/256, 256, 0, stream>>>(hist, zn);

  int total = Bn*NPER;
  gp_hist     <<<(total + 255)/256, 256, 0, stream>>>(scores, hist);
  gp_scan     <<<1, 32, 0, stream>>>(hist, tbin);
  gp_gather   <<<(total + 255)/256, 256, 0, stream>>>(scores, tbin, gcnt, gkeys);
  gp_sortsel  <<<Bn, 1024, 0, stream>>>(gkeys, gcnt, topi, tops);
  gp_transform<<<(Bn*PRE + 255)/256, 256, 0, stream>>>(topi, deltas, iminfo, anchors, props, valid);
  gp_mask     <<<Bn*16, 256, 0, stream>>>(props, maskm);
  gp_nmsscan  <<<Bn, 64, 0, stream>>>(valid, maskm, keep);
  gp_output   <<<Bn, 1024, 0, stream>>>(props, tops, keep, out);
}